// SAE_37495064494602
// MI455X (gfx1250) — compile-verified
//
#include <hip/hip_runtime.h>

typedef __attribute__((ext_vector_type(2))) float        v2f;
typedef __attribute__((ext_vector_type(8))) float        v8f;
typedef __attribute__((ext_vector_type(4))) unsigned int v4u;
typedef __attribute__((ext_vector_type(8))) int          v8i;
typedef __attribute__((ext_vector_type(4))) int          v4i;

#define H_DIM 768
#define M_DIM 4096
#define KTOP  64

#define BLK_COLS 512   // dictionary rows (logit cols) staged per block
#define KCHUNK   16    // K floats per staged chunk
#define LDSROW   18    // 16 + 2 pad dwords (TDM pad: interval=16dw, amount=2dw)

#if __has_builtin(__builtin_amdgcn_tensor_load_to_lds)
#define HAVE_TDM 1
#else
#define HAVE_TDM 0
#endif

// ---------------------------------------------------------------------------
// TDM: 2D tile of enc -> LDS.  tile_dim0=KCHUNK contiguous K floats,
// tile_dim1=BLK_COLS dictionary rows, row stride H_DIM, 4B elements,
// LDS padding 2 dwords every 16 dwords (bank spread for strided WMMA reads).
// D# packing per CDNA5 ISA.
// 6-arg builtin form (clang-23 / therock-10.0 lane).
// ---------------------------------------------------------------------------
#if HAVE_TDM
__device__ __forceinline__ void tdm_load_tile(unsigned lds_off, const float* gsrc) {
    const unsigned long long ga = (unsigned long long)(uintptr_t)gsrc;
    v4u g0;
    g0.x = 1u;                                            // count=1 (valid), user mode
    g0.y = lds_off;                                       // lds_addr (bytes)
    g0.z = (unsigned)(ga & 0xFFFFFFFFu);                  // global_addr[31:0]
    g0.w = (unsigned)((ga >> 32) & 0x1FFFFFFu) | (2u << 30); // addr[56:32] | type=2
    v8i g1;
    g1[0] = (int)((2u << 16)     // data_size = 4B
                | (1u << 20)     // pad_enable
                | (3u << 22)     // pad_interval code 3 -> every 16 dwords
                | (1u << 25));   // pad_amount  code 1 -> 2 dwords
    g1[1] = (int)((unsigned)KCHUNK << 16);    // tensor_dim0[15:0]
    g1[2] = (int)((unsigned)BLK_COLS << 16);  // tensor_dim0 hi=0 | tensor_dim1[15:0]
    g1[3] = (int)((unsigned)KCHUNK << 16);    // tensor_dim1 hi=0 | tile_dim0
    g1[4] = (int)BLK_COLS;                    // tile_dim1 | tile_dim2=0
    g1[5] = (int)H_DIM;                       // tensor_dim0_stride[31:0]
    g1[6] = 0;                                // stride hi | dim1_stride lo (unused)
    g1[7] = 0;
    v4i gz4 = {0, 0, 0, 0};
    v8i gz8 = {0, 0, 0, 0, 0, 0, 0, 0};
    __builtin_amdgcn_tensor_load_to_lds(g0, g1, gz4, gz4, gz8, 0);
}
#endif

// ---------------------------------------------------------------------------
// Encoder GEMM: V_WMMA_F32_16X16X4_F32, B staged in LDS by the Tensor Data
// Mover (double buffered).  Block = 4 waves, each wave 16 rows x 128 cols.
// logits = (x - bias_pre) @ enc^T + bias_enc ; relu ; store dense into z_n.
// ---------------------------------------------------------------------------
__global__ __launch_bounds__(128) void sae_encode_wmma(
    const float* __restrict__ x,        // [N, H]
    const float* __restrict__ enc,      // [M, H]
    const float* __restrict__ bias_pre, // [H]
    const float* __restrict__ bias_enc, // [M]
    float* __restrict__ z)              // [N, M] (z_n region of d_out)
{
    __shared__ float tileB[2][BLK_COLS * LDSROW];   // 2 x 36864 B

    const int lane = threadIdx.x & 31;
    const int wave = threadIdx.x >> 5;
    const int half = lane >> 4;    // 0: K pair (k,k+1); 1: (k+2,k+3)
    const int l16  = lane & 15;

    const int rowBase = blockIdx.y * 16;
    const int colBase = blockIdx.x * BLK_COLS;

    v8f acc[8];
    #pragma unroll
    for (int t = 0; t < 8; ++t)
        #pragma unroll
        for (int v = 0; v < 8; ++v) acc[t][v] = 0.0f;

    const float* xrow = x + (size_t)(rowBase + l16) * H_DIM;
    const float* encBlk = enc + (size_t)colBase * H_DIM;

    const int NCH = H_DIM / KCHUNK;  // 48 chunks

#if HAVE_TDM
    if (wave == 0)  // uniform branch: one TDM issue per workgroup
        tdm_load_tile((unsigned)(uintptr_t)&tileB[0][0], encBlk);
#endif

    for (int ch = 0; ch < NCH; ++ch) {
        const int kb = ch * KCHUNK;

#if HAVE_TDM
        if (wave == 0) {
            if (ch + 1 < NCH) {
                tdm_load_tile((unsigned)(uintptr_t)&tileB[(ch + 1) & 1][0],
                              encBlk + (kb + KCHUNK));
                __builtin_amdgcn_s_wait_tensorcnt((short)1);  // oldest (current buf) done
            } else {
                __builtin_amdgcn_s_wait_tensorcnt((short)0);
            }
        }
        __syncthreads();  // publish tileB[ch&1] to all waves
#else
        // Fallback: cooperative synchronous staging with the same padded layout.
        for (int i = threadIdx.x; i < BLK_COLS * KCHUNK; i += 128) {
            const int r = i >> 4, c = i & 15;
            tileB[ch & 1][r * LDSROW + c] = encBlk[(size_t)r * H_DIM + kb + c];
        }
        __syncthreads();
#endif

        const float* bbuf = &tileB[ch & 1][0];
        #pragma unroll
        for (int kl = 0; kl < KCHUNK; kl += 4) {
            const int kk = kb + kl + 2 * half;
            const int c  = kl + 2 * half;
            v2f bp = *(const v2f*)(bias_pre + kk);
            v2f a  = *(const v2f*)(xrow + kk);
            a.x -= bp.x;
            a.y -= bp.y;
            #pragma unroll
            for (int t = 0; t < 8; ++t) {
                const int r = wave * 128 + t * 16 + l16;
                v2f b = *(const v2f*)(bbuf + r * LDSROW + c);
                acc[t] = __builtin_amdgcn_wmma_f32_16x16x4_f32(
                    false, a, false, b, (short)0, acc[t], false, false);
            }
        }
        __syncthreads();  // all readers done before buffer is re-filled
    }

    #pragma unroll
    for (int t = 0; t < 8; ++t) {
        const int col = colBase + wave * 128 + t * 16 + l16;
        const float be = bias_enc[col];
        #pragma unroll
        for (int v = 0; v < 8; ++v) {
            const int row = rowBase + v + 8 * half;
            float val = acc[t][v] + be;
            val = fmaxf(val, 0.0f);
            z[(size_t)row * M_DIM + col] = val;
        }
    }
}

// ---------------------------------------------------------------------------
// Exact top-64 per row of 4096 via radix binary search on float bits.
// relu => values >= 0 => uint bit pattern is order-preserving.
// Ties at the threshold broken by lowest index (matches jax top_k effect).
// ---------------------------------------------------------------------------
__global__ __launch_bounds__(256) void sae_topk(
    float* __restrict__ z,        // [N, M]
    int*   __restrict__ out_idx,  // [N, 64]
    float* __restrict__ out_val)  // [N, 64]
{
    __shared__ float vals[M_DIM];
    __shared__ int   red[256];
    __shared__ unsigned char flags[M_DIM];

    const int row = blockIdx.x;
    const int tid = threadIdx.x;
    float* zrow = z + (size_t)row * M_DIM;

    for (int i = tid; i < M_DIM; i += 256) vals[i] = zrow[i];
    __syncthreads();

    const int i0 = tid * 16;  // contiguous chunk -> index-ordered compaction

    unsigned T = 0u;
    for (int bit = 31; bit >= 0; --bit) {
        const unsigned cand = T | (1u << bit);
        int c = 0;
        #pragma unroll
        for (int i = 0; i < 16; ++i)
            c += (__float_as_uint(vals[i0 + i]) >= cand) ? 1 : 0;
        red[tid] = c;
        __syncthreads();
        for (int s = 128; s > 0; s >>= 1) {
            if (tid < s) red[tid] += red[tid + s];
            __syncthreads();
        }
        const int total = red[0];
        __syncthreads();
        if (total >= KTOP) T = cand;
    }

    {
        int c = 0;
        #pragma unroll
        for (int i = 0; i < 16; ++i)
            c += (__float_as_uint(vals[i0 + i]) > T) ? 1 : 0;
        red[tid] = c;
        __syncthreads();
        for (int s = 128; s > 0; s >>= 1) {
            if (tid < s) red[tid] += red[tid + s];
            __syncthreads();
        }
    }
    const int cnt_gt = red[0];
    __syncthreads();

    #pragma unroll
    for (int i = 0; i < 16; ++i)
        flags[i0 + i] = (__float_as_uint(vals[i0 + i]) > T) ? 1 : 0;
    __syncthreads();
    if (tid == 0) {  // fill remaining slots with threshold-equal, lowest index first
        int rem = KTOP - cnt_gt;
        for (int i = 0; i < M_DIM && rem > 0; ++i)
            if (__float_as_uint(vals[i]) == T) { flags[i] = 1; --rem; }
    }
    __syncthreads();

    int cnt = 0;
    #pragma unroll
    for (int i = 0; i < 16; ++i) cnt += flags[i0 + i];
    red[tid] = cnt;
    __syncthreads();
    if (tid == 0) {
        int a = 0;
        for (int i = 0; i < 256; ++i) { const int t = red[i]; red[i] = a; a += t; }
    }
    __syncthreads();

    int off = red[tid];
    for (int i = 0; i < 16; ++i) {
        const int gi = i0 + i;
        if (flags[gi]) {
            out_idx[(size_t)row * KTOP + off] = gi;
            out_val[(size_t)row * KTOP + off] = vals[gi];
            ++off;
        }
        zrow[gi] = flags[gi] ? vals[gi] : 0.0f;
    }
}

// ---------------------------------------------------------------------------
// Sparse decode: x_tgt[n,:] = sum_j val_j * enc[idx_j,:] + bias_pre.
// Fused squared-error partial per row (enc is L2-resident: 12.6 MB << 192 MB).
// ---------------------------------------------------------------------------
__global__ __launch_bounds__(256) void sae_decode(
    const float* __restrict__ enc,       // [M, H]
    const float* __restrict__ zL,        // [N, H]
    const float* __restrict__ bias_pre,  // [H]
    const int*   __restrict__ in_idx,    // [N, 64]
    const float* __restrict__ in_val,    // [N, 64]
    float* __restrict__ x_tgt,           // [N, H]
    float* __restrict__ partial)         // [N]
{
    __shared__ float red[256];
    const int row = blockIdx.x;
    const int tid = threadIdx.x;

    float a0 = 0.0f, a1 = 0.0f, a2 = 0.0f;
    const size_t pbase = (size_t)row * KTOP;
    for (int j = 0; j < KTOP; ++j) {
        const int   idx = in_idx[pbase + j];
        const float v   = in_val[pbase + j];
        const float* er = enc + (size_t)idx * H_DIM;
        a0 = fmaf(v, er[tid],       a0);
        a1 = fmaf(v, er[tid + 256], a1);
        a2 = fmaf(v, er[tid + 512], a2);
    }
    a0 += bias_pre[tid];
    a1 += bias_pre[tid + 256];
    a2 += bias_pre[tid + 512];

    const size_t base = (size_t)row * H_DIM;
    x_tgt[base + tid]       = a0;
    x_tgt[base + tid + 256] = a1;
    x_tgt[base + tid + 512] = a2;

    const float d0 = a0 - zL[base + tid];
    const float d1 = a1 - zL[base + tid + 256];
    const float d2 = a2 - zL[base + tid + 512];
    red[tid] = d0 * d0 + d1 * d1 + d2 * d2;
    __syncthreads();
    for (int s = 128; s > 0; s >>= 1) {
        if (tid < s) red[tid] += red[tid + s];
        __syncthreads();
    }
    if (tid == 0) partial[row] = red[0];
}

// ---------------------------------------------------------------------------
// Deterministic final reduction: loss = sum(partial) / (N*H)
// ---------------------------------------------------------------------------
__global__ __launch_bounds__(256) void sae_reduce(
    const float* __restrict__ partial, int n, float* __restrict__ loss)
{
    __shared__ float red[256];
    const int tid = threadIdx.x;
    float s = 0.0f;
    for (int i = tid; i < n; i += 256) s += partial[i];
    red[tid] = s;
    __syncthreads();
    for (int k = 128; k > 0; k >>= 1) {
        if (tid < k) red[tid] += red[tid + k];
        __syncthreads();
    }
    if (tid == 0) loss[0] = red[0] / ((float)n * (float)H_DIM);
}

extern "C" void kernel_launch(void* const* d_in, const int* in_sizes, int n_in,
                              void* d_out, int out_size, void* d_ws, size_t ws_size,
                              hipStream_t stream) {
    const float* zL       = (const float*)d_in[0];  // [B,D,L,H] -> [N,H]
    const float* enc      = (const float*)d_in[1];  // [M,H]
    // d_in[2] = dictionary_dec = enc^T; dec^T == enc, so decode reuses enc rows.
    const float* bias_pre = (const float*)d_in[3];  // [H]
    const float* bias_enc = (const float*)d_in[4];  // [M]

    const int N = in_sizes[0] / H_DIM;  // 16384

    float* out   = (float*)d_out;
    float* loss  = out;                          // [1]
    float* x_tgt = out + 1;                      // [N, H]
    float* z     = out + 1 + (size_t)N * H_DIM;  // [N, M]

    int*   ws_idx     = (int*)d_ws;                                        // [N,64]
    float* ws_val     = (float*)((char*)d_ws + (size_t)N * KTOP * 4);      // [N,64]
    float* ws_partial = (float*)((char*)d_ws + 2 * (size_t)N * KTOP * 4);  // [N]

    dim3 gridEnc(M_DIM / BLK_COLS, N / 16);
    sae_encode_wmma<<<gridEnc, 128, 0, stream>>>(zL, enc, bias_pre, bias_enc, z);
    sae_topk<<<N, 256, 0, stream>>>(z, ws_idx, ws_val);
    sae_decode<<<N, 256, 0, stream>>>(enc, zL, bias_pre, ws_idx, ws_val, x_tgt, ws_partial);
    sae_reduce<<<1, 256, 0, stream>>>(ws_partial, N, loss);
}